// SNNlifadex_48137993453825
// MI455X (gfx1250) — compile-verified
//
#include <hip/hip_runtime.h>

// ---------------------------------------------------------------------------
// LIF-AdEx recurrent SNN on gfx1250 (MI455X).
// Latency-bound sequential recurrence; all GEMMs via V_WMMA_F32_16X16X4_F32
// (exact f32). 4 independent accumulators break the WMMA RAW chain; the
// readout GEMM for step t-1 is fused into step t (halves launch count).
// ---------------------------------------------------------------------------

typedef float v2f __attribute__((ext_vector_type(2)));
typedef float v8f __attribute__((ext_vector_type(8)));

#define T_STEPS 500
#define B_DIM   128
#define F_DIM   128
#define H_DIM   512
#define O_DIM   32

// norse LIFAdEx constants (DT folded in)
#define DT_TAU_MEM   0.1f      // DT * TAU_MEM_INV
#define DT_TAU_SYN   0.2f      // DT * TAU_SYN_INV
#define DT_TAU_ADA   0.002f    // DT * TAU_ADA_INV
#define ADAPT_CUR    4.0f
#define ADAPT_SPK    0.02f
#define DELTA_T      0.5f
#define V_TH         1.0f
#define K_FILTER     0.2231435511314f  // DT * TAU_FILTER_INV

#define WMMA_F32(A, B, C) \
    __builtin_amdgcn_wmma_f32_16x16x4_f32(false, (A), false, (B), (short)0, (C), false, false)

// ---------------------------------------------------------------------------
// 16x16 output tile GEMM slice over K with 4 independent accumulators.
// arow/brow point at row (m=lm) of A and row (n=lm) of B^T respectively;
// both are contiguous-in-K so fragments are 8B float2 loads.
// ---------------------------------------------------------------------------
__device__ __forceinline__ v8f gemm_tile_k(const float* __restrict__ arow,
                                           const float* __restrict__ brow,
                                           int K, int hi)
{
    v8f a0 = {}, a1 = {}, a2 = {}, a3 = {};
    const int kofs = 2 * hi;
    for (int k = 0; k < K; k += 16) {
        v2f af0 = *(const v2f*)(arow + k + kofs);
        v2f bf0 = *(const v2f*)(brow + k + kofs);
        v2f af1 = *(const v2f*)(arow + k + 4 + kofs);
        v2f bf1 = *(const v2f*)(brow + k + 4 + kofs);
        v2f af2 = *(const v2f*)(arow + k + 8 + kofs);
        v2f bf2 = *(const v2f*)(brow + k + 8 + kofs);
        v2f af3 = *(const v2f*)(arow + k + 12 + kofs);
        v2f bf3 = *(const v2f*)(brow + k + 12 + kofs);
        a0 = WMMA_F32(af0, bf0, a0);
        a1 = WMMA_F32(af1, bf1, a1);
        a2 = WMMA_F32(af2, bf2, a2);
        a3 = WMMA_F32(af3, bf3, a3);
    }
    return (a0 + a1) + (a2 + a3);
}

// ---------------------------------------------------------------------------
// Zero the persistent state (v, i, a) and the t=0 "previous spikes" buffer.
// ---------------------------------------------------------------------------
__global__ __launch_bounds__(256)
void snn_init(float* __restrict__ z0, float* __restrict__ v_s,
              float* __restrict__ i_s, float* __restrict__ a_s)
{
    int tid = blockIdx.x * 256 + threadIdx.x;   // 0 .. B*H-1
    z0[tid] = 0.0f;
    v_s[tid] = 0.0f;
    i_s[tid] = 0.0f;
    a_s[tid] = 0.0f;
}

// ---------------------------------------------------------------------------
// One recurrence step.  Grid: 32 blocks x 256 threads (8 waves).
//   block = (btile 0..7) x (hblk 0..3); wave w owns the 16x16 tile at
//   rows b0..b0+15 (M = batch), cols h0..h0+15 (N = hidden).
// Computes  r = x[t] @ W_in^T + z_prev @ W_rec^T  (K = 128 + 512), then the
// pointwise AdEx update, spike threshold, state write-back, z_new store.
// Additionally, wave 0 of blocks 0..15 computes one 16x16 tile of the
// PREVIOUS step's readout  y[t-1] = z_prev @ W_out^T + b_out  (fused; z_prev
// is globally visible because it was written by the previous kernel launch).
// ---------------------------------------------------------------------------
__global__ __launch_bounds__(256)
void snn_step(const float* __restrict__ x_t,     // (B, F)  = x + t*B*F
              const float* __restrict__ W_in,    // (H, F)
              const float* __restrict__ W_rec,   // (H, H)
              const float* __restrict__ W_out,   // (O, H)
              const float* __restrict__ b_out,   // (O)
              const float* __restrict__ z_prev,  // (B, H)
              float*       __restrict__ z_new,   // (B, H)
              float*       __restrict__ y_prev,  // (B, O) for t-1, or nullptr
              float* __restrict__ v_s,
              float* __restrict__ i_s,
              float* __restrict__ a_s)
{
    const int wave = threadIdx.x >> 5;
    const int lane = threadIdx.x & 31;
    const int hi   = lane >> 4;       // half-wave select
    const int lm   = lane & 15;       // M (for A) / N (for B) lane index

    const int b0 = (blockIdx.x & 7) * 16;                 // M tile base (batch)
    const int h0 = (blockIdx.x >> 3) * 128 + wave * 16;   // N tile base (hidden)

    // ---- r = x[t] @ W_in^T + z_prev @ W_rec^T for this 16x16 tile ----
    v8f acc = gemm_tile_k(x_t    + (size_t)(b0 + lm) * F_DIM,
                          W_in   + (size_t)(h0 + lm) * F_DIM, F_DIM, hi)
            + gemm_tile_k(z_prev + (size_t)(b0 + lm) * H_DIM,
                          W_rec  + (size_t)(h0 + lm) * H_DIM, H_DIM, hi);

    // ---- pointwise AdEx update (8 elems / lane, C-layout of the tile) ----
    #pragma unroll
    for (int j = 0; j < 8; ++j) {
        const int m   = j + 8 * hi;                      // row within tile
        const int idx = (b0 + m) * H_DIM + (h0 + lm);    // (b, h) flat

        const float v_old = v_s[idx];
        const float i_old = i_s[idx];
        const float a_old = a_s[idx];

        const float dv = DT_TAU_MEM * ((-v_old)
                         + DELTA_T * __expf((v_old - V_TH) * 2.0f)
                         + i_old - a_old);
        const float v_dec = v_old + dv;
        const float i_dec = i_old - DT_TAU_SYN * i_old;
        const float a_dec = a_old + DT_TAU_ADA * (ADAPT_CUR * v_old - a_old);

        const float z = (v_dec - V_TH > 0.0f) ? 1.0f : 0.0f;

        v_s[idx]   = (1.0f - z) * v_dec;        // V_RESET = 0
        i_s[idx]   = i_dec + acc[j];            // + ic + z_prev @ W_rec^T
        a_s[idx]   = a_dec + z * ADAPT_SPK;
        z_new[idx] = z;
    }

    // ---- fused readout of the PREVIOUS step: y[t-1] = z_prev @ W_out^T ----
    // 16 tiles (8 M x 2 N) handled by wave 0 of blocks 0..15. Wave-uniform
    // branch: EXEC stays all-ones inside, as WMMA requires.
    if (y_prev != nullptr && blockIdx.x < 16 && wave == 0) {
        const int bm0 = (blockIdx.x & 7) * 16;     // M tile (batch)
        const int o0  = (blockIdx.x >> 3) * 16;    // N tile (output)
        v8f yacc = gemm_tile_k(z_prev + (size_t)(bm0 + lm) * H_DIM,
                               W_out  + (size_t)(o0  + lm) * H_DIM, H_DIM, hi);
        #pragma unroll
        for (int j = 0; j < 8; ++j) {
            const int m = j + 8 * hi;
            y_prev[(bm0 + m) * O_DIM + (o0 + lm)] = yacc[j] + b_out[o0 + lm];
        }
    }
}

// ---------------------------------------------------------------------------
// Standalone readout (used once, for the final step's spikes).
// M=B=128, N=O=32, K=H=512.  One block, 8 waves; wave w owns M-tile w and
// loops over the 2 N-tiles.
// ---------------------------------------------------------------------------
__global__ __launch_bounds__(256)
void snn_readout(const float* __restrict__ z,      // (B, H)
                 const float* __restrict__ W_out,  // (O, H)
                 const float* __restrict__ b_out,  // (O)
                 float*       __restrict__ y_t)    // (B, O)
{
    const int wave = threadIdx.x >> 5;
    const int lane = threadIdx.x & 31;
    const int hi   = lane >> 4;
    const int lm   = lane & 15;

    const int b0 = wave * 16;
    const float* zrow = z + (size_t)(b0 + lm) * H_DIM;

    #pragma unroll
    for (int nt = 0; nt < 2; ++nt) {
        const int o0 = nt * 16;
        v8f acc = gemm_tile_k(zrow, W_out + (size_t)(o0 + lm) * H_DIM, H_DIM, hi);
        #pragma unroll
        for (int j = 0; j < 8; ++j) {
            const int m = j + 8 * hi;
            y_t[(b0 + m) * O_DIM + (o0 + lm)] = acc[j] + b_out[o0 + lm];
        }
    }
}

// ---------------------------------------------------------------------------
// Exponential low-pass filter over T.  One thread per (b, o); reads of
// y[t*4096 + tid] are fully coalesced per time step.
// ---------------------------------------------------------------------------
__global__ __launch_bounds__(256)
void snn_filter(const float* __restrict__ y, float* __restrict__ out)
{
    const int tid = blockIdx.x * 256 + threadIdx.x;   // 0 .. B*O-1 (4096)
    float o = y[tid];
    out[tid] = o;
    for (int t = 1; t < T_STEPS; ++t) {
        const float yt = y[(size_t)t * (B_DIM * O_DIM) + tid];
        o += K_FILTER * (yt - o);
        out[(size_t)t * (B_DIM * O_DIM) + tid] = o;
    }
}

// ---------------------------------------------------------------------------
// Host-side orchestration (graph-capture safe: only kernel launches).
// Workspace layout (floats):
//   [0]          z ping-pong buf 0      (B*H = 65536)
//   [65536]      z ping-pong buf 1
//   [2*65536]    v state
//   [3*65536]    i state
//   [4*65536]    a state
//   [5*65536]    y  (T*B*O = 2,048,000)
// Total ~9.5 MB.  ~503 kernel launches per call.
// ---------------------------------------------------------------------------
extern "C" void kernel_launch(void* const* d_in, const int* in_sizes, int n_in,
                              void* d_out, int out_size, void* d_ws, size_t ws_size,
                              hipStream_t stream)
{
    const float* x     = (const float*)d_in[0];  // (T, B, F)
    const float* W_in  = (const float*)d_in[1];  // (H, F)
    const float* W_rec = (const float*)d_in[2];  // (H, H)
    const float* W_out = (const float*)d_in[3];  // (O, H)
    const float* b_out = (const float*)d_in[4];  // (O)

    float* ws   = (float*)d_ws;
    const size_t SB = (size_t)B_DIM * H_DIM;     // 65536
    float* zbuf0 = ws;
    float* zbuf1 = ws + SB;
    float* v_s   = ws + 2 * SB;
    float* i_s   = ws + 3 * SB;
    float* a_s   = ws + 4 * SB;
    float* y     = ws + 5 * SB;                  // (T, B, O)

    // zero state + z(t=-1) every call (deterministic, no carried state)
    snn_init<<<SB / 256, 256, 0, stream>>>(zbuf0, v_s, i_s, a_s);

    float* zpp[2] = { zbuf0, zbuf1 };
    for (int t = 0; t < T_STEPS; ++t) {
        const float* zp = zpp[t & 1];
        float*       zn = zpp[(t + 1) & 1];
        float* yprev = (t == 0) ? nullptr : y + (size_t)(t - 1) * B_DIM * O_DIM;
        snn_step<<<32, 256, 0, stream>>>(x + (size_t)t * B_DIM * F_DIM,
                                         W_in, W_rec, W_out, b_out,
                                         zp, zn, yprev, v_s, i_s, a_s);
    }
    // readout for the last step's spikes (z[T-1] lives in zpp[T & 1])
    snn_readout<<<1, 256, 0, stream>>>(zpp[T_STEPS & 1], W_out, b_out,
                                       y + (size_t)(T_STEPS - 1) * B_DIM * O_DIM);

    snn_filter<<<(B_DIM * O_DIM) / 256, 256, 0, stream>>>(y, (float*)d_out);
}